// GCNEncoder_22660247453755
// MI455X (gfx1250) — compile-verified
//
#include <hip/hip_runtime.h>
#include <hip/hip_bf16.h>

typedef float v2f __attribute__((ext_vector_type(2)));
typedef float v8f __attribute__((ext_vector_type(8)));

#define DFEAT 128

// ---------------------------------------------------------------------------
// Generic zero-fill (grid-stride)
// ---------------------------------------------------------------------------
__global__ void gcn_zero(float* __restrict__ p, int n) {
    int i = blockIdx.x * blockDim.x + threadIdx.x;
    int stride = gridDim.x * blockDim.x;
    for (; i < n; i += stride) p[i] = 0.0f;
}

// ---------------------------------------------------------------------------
// Degree accumulation: deg_out[src[e]] += 1 ; deg_in[dst[e]] += 1
// (accumulated directly into the norm buffers, transformed afterwards)
// ---------------------------------------------------------------------------
__global__ void gcn_degrees(const int* __restrict__ src, const int* __restrict__ dst,
                            float* __restrict__ deg_out, float* __restrict__ deg_in,
                            int n_edges) {
    int e = blockIdx.x * blockDim.x + threadIdx.x;
    if (e >= n_edges) return;
    atomicAdd(deg_out + src[e], 1.0f);
    atomicAdd(deg_in + dst[e], 1.0f);
}

// norm = rsqrt(max(deg, 1))  (in place, both buffers)
__global__ void gcn_norms(float* __restrict__ norm_src, float* __restrict__ norm_dst, int n) {
    int i = blockIdx.x * blockDim.x + threadIdx.x;
    if (i >= n) return;
    norm_src[i] = rsqrtf(fmaxf(norm_src[i], 1.0f));
    norm_dst[i] = rsqrtf(fmaxf(norm_dst[i], 1.0f));
}

// ---------------------------------------------------------------------------
// Edge scatter with both norms folded in (row scaling commutes through the
// GEMM, so diag(norm_dst) can be applied here instead of in the GEMM):
//   m[dst] += x[src] * norm_src[src] * norm_dst[dst]
// One wave32 per edge; each lane moves float4 (32 lanes * 4 = 128 floats).
// The extra multiply is free: this kernel is atomic/memory bound.
// ---------------------------------------------------------------------------
__global__ void gcn_scatter(const float* __restrict__ X, const float* __restrict__ norm_src,
                            const float* __restrict__ norm_dst,
                            const int* __restrict__ src, const int* __restrict__ dst,
                            float* __restrict__ M, int n_edges) {
    int wid  = (blockIdx.x * blockDim.x + threadIdx.x) >> 5;
    int lane = threadIdx.x & 31;
    if (wid >= n_edges) return;
    int s = src[wid];
    int t = dst[wid];
    float sc = norm_src[s] * norm_dst[t];
    const float4* xp = (const float4*)(X + (size_t)s * DFEAT);
    float4 v = xp[lane];
    float* mp = M + (size_t)t * DFEAT + lane * 4;
    atomicAdd(mp + 0, v.x * sc);
    atomicAdd(mp + 1, v.y * sc);
    atomicAdd(mp + 2, v.z * sc);
    atomicAdd(mp + 3, v.w * sc);
}

// ---------------------------------------------------------------------------
// H = relu(M @ W + bias), W is DxD row-major. (Norms already folded into M.)
// One wave per 16x16 output tile, K-loop of V_WMMA_F32_16X16X4_F32.
//
// A (16x4 f32) layout: lanes 0-15 -> M=lane, VGPR0=K0, VGPR1=K1;
//                      lanes 16-31 -> M=lane-16, VGPR0=K2, VGPR1=K3.
// B (4x16 f32) mirrors with N = lane%16.
// C/D (16x16 f32, 8 VGPRs): VGPR r -> row r (lanes 0-15) / row r+8 (lanes 16-31),
//                           col = lane%16.
// ---------------------------------------------------------------------------
__global__ void gcn_gemm_wmma(const float* __restrict__ M,
                              const float* __restrict__ W, const float* __restrict__ bias,
                              float* __restrict__ H, int n_rows) {
    const int n_col_tiles = DFEAT / 16;  // 8
    int wave_id = (blockIdx.x * blockDim.x + threadIdx.x) >> 5;
    int lane    = threadIdx.x & 31;
    int row_tile = wave_id / n_col_tiles;
    int col_tile = wave_id % n_col_tiles;
    int r0 = row_tile * 16;
    int c0 = col_tile * 16;
    if (r0 >= n_rows) return;  // wave-uniform: EXEC stays all-ones for live waves

    int lm = lane & 15;            // row (A) / col (B,C,D) within tile
    int kg = (lane >> 4) << 1;     // K sub-offset: 0 (lanes 0-15) or 2 (lanes 16-31)

    const float* arowp = M + (size_t)(r0 + lm) * DFEAT;
    int ncol = c0 + lm;

    v8f acc = {};
    #pragma unroll 8
    for (int k = 0; k < DFEAT; k += 4) {
        v2f a, b;
        a.x = arowp[k + kg];
        a.y = arowp[k + kg + 1];
        b.x = W[(size_t)(k + kg)     * DFEAT + ncol];
        b.y = W[(size_t)(k + kg + 1) * DFEAT + ncol];
        acc = __builtin_amdgcn_wmma_f32_16x16x4_f32(
            /*neg_a=*/false, a, /*neg_b=*/false, b,
            /*c_mod=*/(short)0, acc, /*reuse_a=*/false, /*reuse_b=*/false);
    }

    float bv = bias[ncol];
    int mbase = r0 + ((lane >> 4) << 3);  // lanes 16-31 handle rows +8
    #pragma unroll
    for (int r = 0; r < 8; ++r) {
        float v = acc[r] + bv;
        H[(size_t)(mbase + r) * DFEAT + ncol] = fmaxf(v, 0.0f);
    }
}

// ---------------------------------------------------------------------------
// Pooling: out[g] += h[i] for g = graph_ids[i]; counts[g] += 1. Wave per node.
// ---------------------------------------------------------------------------
__global__ void gcn_pool_accum(const float* __restrict__ H, const int* __restrict__ gid,
                               float* __restrict__ out, float* __restrict__ counts,
                               int n_nodes) {
    int wid  = (blockIdx.x * blockDim.x + threadIdx.x) >> 5;
    int lane = threadIdx.x & 31;
    if (wid >= n_nodes) return;
    int g = gid[wid];
    const float4* hp = (const float4*)(H + (size_t)wid * DFEAT);
    float4 v = hp[lane];
    float* op = out + (size_t)g * DFEAT + lane * 4;
    atomicAdd(op + 0, v.x);
    atomicAdd(op + 1, v.y);
    atomicAdd(op + 2, v.z);
    atomicAdd(op + 3, v.w);
    if (lane == 0) atomicAdd(counts + g, 1.0f);
}

__global__ void gcn_pool_div(float* __restrict__ out, const float* __restrict__ counts,
                             int total) {
    int i = blockIdx.x * blockDim.x + threadIdx.x;
    if (i >= total) return;
    out[i] /= fmaxf(counts[i / DFEAT], 1.0f);
}

// ---------------------------------------------------------------------------
// Launch
// ---------------------------------------------------------------------------
extern "C" void kernel_launch(void* const* d_in, const int* in_sizes, int n_in,
                              void* d_out, int out_size, void* d_ws, size_t ws_size,
                              hipStream_t stream) {
    const float* node_feats = (const float*)d_in[0];
    const float* W1         = (const float*)d_in[1];
    const float* b1         = (const float*)d_in[2];
    const float* W2         = (const float*)d_in[3];
    const float* b2         = (const float*)d_in[4];
    const int*   src        = (const int*)d_in[5];
    const int*   dst        = (const int*)d_in[6];
    const int*   graph_ids  = (const int*)d_in[7];

    const int n_nodes  = in_sizes[7];           // 50000
    const int n_edges  = in_sizes[5];           // 600000
    const int n_graphs = out_size / DFEAT;      // 500

    // Workspace layout (floats)
    float* ws       = (float*)d_ws;
    float* norm_src = ws;                        // n
    float* norm_dst = ws + n_nodes;              // n
    float* mbuf     = ws + 2 * (size_t)n_nodes;                       // n*D
    float* hbuf     = mbuf + (size_t)n_nodes * DFEAT;                 // n*D
    float* counts   = hbuf + (size_t)n_nodes * DFEAT;                 // n_graphs
    float* outf     = (float*)d_out;

    const int ZB = 256;
    const int nd_elems = n_nodes * DFEAT;

    // 1) degrees -> norms
    gcn_zero<<<1024, ZB, 0, stream>>>(norm_src, 2 * n_nodes);
    gcn_degrees<<<(n_edges + ZB - 1) / ZB, ZB, 0, stream>>>(src, dst, norm_src, norm_dst, n_edges);
    gcn_norms<<<(n_nodes + ZB - 1) / ZB, ZB, 0, stream>>>(norm_src, norm_dst, n_nodes);

    const int scatter_blocks = (n_edges * 32 + ZB - 1) / ZB;
    const int gemm_waves     = (n_nodes / 16) * (DFEAT / 16);   // 16 | n_nodes
    const int gemm_blocks    = (gemm_waves * 32 + ZB - 1) / ZB;

    // 2) layer 1: m = scatter(node_feats, both norms); h = relu(m @ W1 + b1)
    gcn_zero<<<4096, ZB, 0, stream>>>(mbuf, nd_elems);
    gcn_scatter<<<scatter_blocks, ZB, 0, stream>>>(node_feats, norm_src, norm_dst, src, dst, mbuf, n_edges);
    gcn_gemm_wmma<<<gemm_blocks, ZB, 0, stream>>>(mbuf, W1, b1, hbuf, n_nodes);

    // 3) layer 2: m = scatter(h, both norms); h = relu(m @ W2 + b2)
    gcn_zero<<<4096, ZB, 0, stream>>>(mbuf, nd_elems);
    gcn_scatter<<<scatter_blocks, ZB, 0, stream>>>(hbuf, norm_src, norm_dst, src, dst, mbuf, n_edges);
    gcn_gemm_wmma<<<gemm_blocks, ZB, 0, stream>>>(mbuf, W2, b2, hbuf, n_nodes);

    // 4) pooling: per-graph mean
    gcn_zero<<<(out_size + ZB - 1) / ZB, ZB, 0, stream>>>(outf, out_size);
    gcn_zero<<<(n_graphs + ZB - 1) / ZB, ZB, 0, stream>>>(counts, n_graphs);
    gcn_pool_accum<<<(n_nodes * 32 + ZB - 1) / ZB, ZB, 0, stream>>>(hbuf, graph_ids, outf, counts, n_nodes);
    gcn_pool_div<<<(out_size + ZB - 1) / ZB, ZB, 0, stream>>>(outf, counts, out_size);
}